// SingleBoxesOverlapNet_64424509440044
// MI455X (gfx1250) — compile-verified
//
#include <hip/hip_runtime.h>
#include <math.h>

// ---------------------------------------------------------------------------
// Rotated-box pairwise BEV overlap area, MI455X (gfx1250, wave32).
//
// Roofline: 84 KB input (L2-resident), 8 MB output (~0.34us @ 23.3 TB/s),
// 2M independent pair computations -> VALU-bound. No matmul structure, so
// WMMA does not apply; we use the CDNA5 async global->LDS path (ASYNCcnt),
// LDS polygon scratch (ds ops, odd-stride bank-conflict-free), and wave32
// tiling (wave = 1 A-box x 32 B-boxes -> coalesced 128B stores, and the SAT
// reject lets ~84% of waves skip the clip entirely via s_cbranch_execz).
// ---------------------------------------------------------------------------

#define TILE_A   8
#define TILE_B   32
#define NTHREADS 256
#define VSTRIDE  17   // dwords per thread per polygon buffer (odd -> no bank conflicts)

#if defined(__gfx1250__) && __has_builtin(__builtin_amdgcn_global_load_async_to_lds_b32)
#define ASYNC_STAGE 1
typedef __attribute__((address_space(1))) int* gint_ptr;
typedef __attribute__((address_space(3))) int* lint_ptr;
#endif

// Clip convex polygon src (nv verts, (x,y) pairs) against half-plane
// nx*x + ny*y <= d. Writes result to dst, returns new vertex count (<= nv+1).
__device__ __forceinline__ int clip_halfplane(const float* __restrict__ src,
                                              float* __restrict__ dst,
                                              int nv, float nx, float ny, float d)
{
    if (nv == 0) return 0;
    int   nout = 0;
    float xP = src[2 * (nv - 1)], yP = src[2 * (nv - 1) + 1];
    float fP = nx * xP + ny * yP - d;
    for (int j = 0; j < nv; ++j) {
        float xC = src[2 * j], yC = src[2 * j + 1];
        float fC = nx * xC + ny * yC - d;
        bool  inC = (fC <= 0.0f), inP = (fP <= 0.0f);
        if (inC != inP) {                       // edge crosses the plane
            float t = fP / (fP - fC);           // denom != 0 when signs differ
            dst[2 * nout]     = fmaf(t, xC - xP, xP);
            dst[2 * nout + 1] = fmaf(t, yC - yP, yP);
            ++nout;
        }
        if (inC) { dst[2 * nout] = xC; dst[2 * nout + 1] = yC; ++nout; }
        xP = xC; yP = yC; fP = fC;
    }
    return nout;
}

__global__ __launch_bounds__(NTHREADS)
void rotated_overlap_kernel(const float* __restrict__ boxes,   // [m+n][7]
                            float* __restrict__ out,           // [m][n]
                            int m, int n)
{
    __shared__ float rawA[TILE_A * 7];
    __shared__ float rawB[TILE_B * 7];
    __shared__ float aDer[TILE_A][9];   // cx, cy, hx, hy, cos, sin  (stride 9: odd)
    __shared__ float bDer[TILE_B][9];
    __shared__ float polyP[NTHREADS * VSTRIDE];
    __shared__ float polyQ[NTHREADS * VSTRIDE];

    const int tid   = threadIdx.x;
    const int aBase = blockIdx.y * TILE_A;
    const int bBase = blockIdx.x * TILE_B;

    // ---- Stage raw box data for this tile into LDS ------------------------
    {
        int mA = m - aBase; if (mA > TILE_A) mA = TILE_A;
        int mB = n - bBase; if (mB > TILE_B) mB = TILE_B;
        int na = mA * 7;              // <= 56 dwords
        int nb = mB * 7;              // <= 224 dwords
#ifdef ASYNC_STAGE
        if (tid < na)
            __builtin_amdgcn_global_load_async_to_lds_b32(
                (gint_ptr)(boxes + (size_t)aBase * 7 + tid),
                (lint_ptr)(&rawA[tid]), 0, 0);
        if (tid < nb)
            __builtin_amdgcn_global_load_async_to_lds_b32(
                (gint_ptr)(boxes + (size_t)(m + bBase) * 7 + tid),
                (lint_ptr)(&rawB[tid]), 0, 0);
#if __has_builtin(__builtin_amdgcn_s_wait_asynccnt)
        __builtin_amdgcn_s_wait_asynccnt(0);
#else
        asm volatile("s_wait_asynccnt 0" ::: "memory");
#endif
#else
        if (tid < na) rawA[tid] = boxes[(size_t)aBase * 7 + tid];
        if (tid < nb) rawB[tid] = boxes[(size_t)(m + bBase) * 7 + tid];
#endif
    }
    __syncthreads();

    // ---- Per-box derived data (center, half-extents, cos/sin), once -------
    if (tid < TILE_A) {                                   // wave 0
        float x = 0.f, y = 0.f, hx = 0.f, hy = 0.f, c = 1.f, s = 0.f;
        if (aBase + tid < m) {
            const float* p = &rawA[tid * 7];
            x = p[0]; y = p[1]; hx = 0.5f * p[3]; hy = 0.5f * p[4];
            __sincosf(p[6], &s, &c);
        }
        aDer[tid][0] = x; aDer[tid][1] = y; aDer[tid][2] = hx;
        aDer[tid][3] = hy; aDer[tid][4] = c; aDer[tid][5] = s;
    }
    if (tid >= 32 && tid < 32 + TILE_B) {                 // wave 1
        int bt = tid - 32;
        float x = 0.f, y = 0.f, hx = 0.f, hy = 0.f, c = 1.f, s = 0.f;
        if (bBase + bt < n) {
            const float* p = &rawB[bt * 7];
            x = p[0]; y = p[1]; hx = 0.5f * p[3]; hy = 0.5f * p[4];
            __sincosf(p[6], &s, &c);
        }
        bDer[bt][0] = x; bDer[bt][1] = y; bDer[bt][2] = hx;
        bDer[bt][3] = hy; bDer[bt][4] = c; bDer[bt][5] = s;
    }
    __syncthreads();

    // ---- One pair per thread: wave = (1 A-box) x (32 contiguous B-boxes) --
    const int  ai = tid >> 5;          // 0..7   (uniform per wave -> LDS broadcast)
    const int  bi = tid & 31;          // 0..31  (lane id -> conflict-free, stride 9)
    const int  a  = aBase + ai;
    const int  b  = bBase + bi;
    const bool active = (a < m) && (b < n);

    float axc = aDer[ai][0], ayc = aDer[ai][1], ahx = aDer[ai][2],
          ahy = aDer[ai][3], ac  = aDer[ai][4], as_ = aDer[ai][5];
    float bxc = bDer[bi][0], byc = bDer[bi][1], bhx = bDer[bi][2],
          bhy = bDer[bi][3], bc  = bDer[bi][4], bs  = bDer[bi][5];

    // B's center in A's local (axis-aligned) frame
    float tx = bxc - axc, ty = byc - ayc;
    float px =  tx * ac + ty * as_;
    float py = -tx * as_ + ty * ac;
    // relative rotation r = yawB - yawA via angle-subtraction (no per-pair trig)
    float cr = ac * bc + as_ * bs;
    float sr = bs * ac - bc * as_;

    // Exact SAT reject on A's two axes: B's extent projected on A-x / A-y
    float extx = fabsf(bhx * cr) + fabsf(bhy * sr);
    float exty = fabsf(bhx * sr) + fabsf(bhy * cr);
    bool  maybe = active && (fabsf(px) <= ahx + extx) && (fabsf(py) <= ahy + exty);

    float area = 0.0f;
    if (maybe) {   // ~0.5% of lanes; ~84% of waves skip via s_cbranch_execz
        float* P = &polyP[tid * VSTRIDE];
        float* Q = &polyQ[tid * VSTRIDE];

        // B's corners in A's frame, in convex (CCW/CW-consistent) order
        const float ex[4] = {  bhx,  bhx, -bhx, -bhx };
        const float ey[4] = {  bhy, -bhy, -bhy,  bhy };
#pragma unroll
        for (int j = 0; j < 4; ++j) {
            P[2 * j]     = px + ex[j] * cr - ey[j] * sr;
            P[2 * j + 1] = py + ex[j] * sr + ey[j] * cr;
        }

        // Sutherland-Hodgman against A's four half-planes (max 8 verts)
        int nv = 4;
        nv = clip_halfplane(P, Q, nv,  1.f,  0.f, ahx);   //  x <= ahx
        nv = clip_halfplane(Q, P, nv, -1.f,  0.f, ahx);   // -x <= ahx
        nv = clip_halfplane(P, Q, nv,  0.f,  1.f, ahy);   //  y <= ahy
        nv = clip_halfplane(Q, P, nv,  0.f, -1.f, ahy);   // -y <= ahy

        if (nv >= 3) {                                    // shoelace
            float s2 = 0.0f;
            float xp = P[2 * (nv - 1)], yp = P[2 * (nv - 1) + 1];
            for (int j = 0; j < nv; ++j) {
                float xc = P[2 * j], yc = P[2 * j + 1];
                s2 += xp * yc - xc * yp;
                xp = xc; yp = yc;
            }
            area = 0.5f * fabsf(s2);
        }
    }

    if (active)
        out[(size_t)a * n + b] = area;   // coalesced: wave covers 32 contiguous b
}

extern "C" void kernel_launch(void* const* d_in, const int* in_sizes, int n_in,
                              void* d_out, int out_size, void* d_ws, size_t ws_size,
                              hipStream_t stream)
{
    const float* boxes = (const float*)d_in[0];
    float*       out   = (float*)d_out;

    const int total = in_sizes[0] / 7;   // 3000 boxes
    const int m     = total / 3;         // 1000 A-boxes
    const int n     = total - m;         // 2000 B-boxes

    dim3 block(NTHREADS);
    dim3 grid((n + TILE_B - 1) / TILE_B, (m + TILE_A - 1) / TILE_A);
    hipLaunchKernelGGL(rotated_overlap_kernel, grid, block, 0, stream,
                       boxes, out, m, n);
}